// KalmanFilter_78039555769033
// MI455X (gfx1250) — compile-verified
//
#include <hip/hip_runtime.h>
#include <math.h>

typedef float v2f __attribute__((ext_vector_type(2)));
typedef float v8f __attribute__((ext_vector_type(8)));

// problem dims
#define Bc 256
#define Tc 64
#define DZc 64
#define DAc 32
#define Kc 8
#define Hc 64
#define DOBSc 16

// ---- output layout (floats), tuple order of the reference ----
constexpr size_t O_ZF   = 0;
constexpr size_t O_PF   = O_ZF   + (size_t)Bc*Tc*DZc;
constexpr size_t O_ZP   = O_PF   + (size_t)Bc*Tc*DZc*DZc;
constexpr size_t O_PP   = O_ZP   + (size_t)Bc*Tc*DZc;
constexpr size_t O_AF   = O_PP   + (size_t)Bc*Tc*DZc*DZc;
constexpr size_t O_AP   = O_AF   + (size_t)Bc*Tc*DAc;
constexpr size_t O_S    = O_AP   + (size_t)Bc*Tc*DAc;
constexpr size_t O_AL   = O_S    + (size_t)Bc*Tc*DAc*DAc;
constexpr size_t O_AIMM = O_AL   + (size_t)Bc*Tc*Kc;
constexpr size_t O_R    = O_AIMM + (size_t)Bc*Tc*Kc;
constexpr size_t O_Q    = O_R    + (size_t)DAc*DAc;

// ---- LDS layout (float offsets) ----
constexpr int oP    = 0;
constexpr int oPf   = oP    + 64*64;
constexpr int oAmix = oPf   + 64*64;
constexpr int oTt   = oAmix + 64*64;
constexpr int oIKC  = oTt   + 64*64;
constexpr int oQm   = oIKC  + 64*64;
constexpr int oCmix = oQm   + 64*64;
constexpr int oCst  = oCmix + 32*64;
constexpr int oPCm  = oCst  + 32*64;
constexpr int oKg   = oPCm  + 64*32;
constexpr int oSall = oKg   + 64*32;      // 8 x 32 x 33 (padded)
constexpr int oSk   = oSall + 8*32*33;
constexpr int oLinv = oSk   + 32*33;
constexpr int oSinv = oLinv + 32*33;
constexpr int oRm   = oSinv + 32*33;
constexpr int oRall = oRm   + 32*32;      // 8 x 32
constexpr int oAv   = oRall + 8*32;
constexpr int oZv   = oAv   + 32;
constexpr int oZf   = oZv   + 64;
constexpr int oAf   = oZf   + 64;
constexpr int oXb   = oAf   + 32;         // 112
constexpr int oGi   = oXb   + 112;        // 192
constexpr int oGh   = oGi   + 192;        // 192
constexpr int oHv   = oGh   + 192;        // 64
constexpr int oAl   = oHv   + 64;         // 8 (alpha carry)
constexpr int oLl   = oAl   + 8;          // 8
constexpr int oLg   = oLl   + 8;          // 8 (logits)
constexpr int oEnd  = oLg   + 8;
constexpr int SMEM_BYTES = oEnd * 4;      // ~186 KB, fits in 320 KB WGP LDS

// ===========================================================================
// Tiled GEMM over the 8 waves of the block, one 16x16 tile per WMMA chain.
// D = op(A)*op(B) (+ Cinit). f32 WMMA 16x16x4, K stepped by 4.
// Fragment layouts per CDNA5 ISA 7.12.2 (wave32):
//   A 16x4 : lane(0..15)->M, lanes16..31 same M, K = 2*half + {0,1} in v0/v1
//   B 4x16 : VGPR r holds rows {r, r+2}; lanes: N = lane&15, half selects +2
//   C 16x16: VGPR r holds rows {r, r+8}; N = lane&15
// ===========================================================================
__device__ __forceinline__ void mm_wave(
    float* dst, int ldd,
    const float* A, int lda, bool tA,
    const float* Bm, int ldb, bool tB,
    int M, int N, int Kd,
    const float* Cinit, int ldc,
    int wave, int lane)
{
  const int ncol  = N >> 4;
  const int nTile = (M >> 4) * ncol;
  const int half  = lane >> 4;
  const int lid   = lane & 15;
  for (int tIdx = wave; tIdx < nTile; tIdx += 8) {
    const int m0 = (tIdx / ncol) << 4;
    const int n0 = (tIdx % ncol) << 4;
    v8f acc;
    if (Cinit) {
      #pragma unroll
      for (int r = 0; r < 8; ++r)
        acc[r] = Cinit[(size_t)(m0 + r + 8*half)*ldc + n0 + lid];
    } else {
      #pragma unroll
      for (int r = 0; r < 8; ++r) acc[r] = 0.0f;
    }
    for (int kk = 0; kk < Kd; kk += 4) {
      const int ka = kk + 2*half;
      v2f a, b;
      if (!tA) { a.x = A[(m0+lid)*lda + ka];     a.y = A[(m0+lid)*lda + ka + 1]; }
      else     { a.x = A[ka*lda + m0 + lid];     a.y = A[(ka+1)*lda + m0 + lid]; }
      if (!tB) { b.x = Bm[ka*ldb + n0 + lid];    b.y = Bm[(ka+1)*ldb + n0 + lid]; }
      else     { b.x = Bm[(n0+lid)*ldb + ka];    b.y = Bm[(n0+lid)*ldb + ka + 1]; }
      acc = __builtin_amdgcn_wmma_f32_16x16x4_f32(false, a, false, b,
                                                  (short)0, acc, false, false);
    }
    #pragma unroll
    for (int r = 0; r < 8; ++r)
      dst[(size_t)(m0 + r + 8*half)*ldd + n0 + lid] = acc[r];
  }
}

// In-place lower Cholesky of a 32x32 (ld=33) matrix; each wave factors its own
// S (8 in parallel).  All 256 threads execute the uniform loop (barriers).
__device__ __forceinline__ void chol32_block(float* S, int lane)
{
  for (int k = 0; k < 32; ++k) {
    float inv = 1.0f / sqrtf(S[k*33 + k]);
    if (lane == k)      S[k*33 + k] = S[k*33 + k] * inv;   // = sqrt(d)
    else if (lane > k)  S[lane*33 + k] *= inv;
    __syncthreads();
    if (lane > k) {
      float lk = S[lane*33 + k];
      for (int m = k + 1; m <= lane; ++m)
        S[lane*33 + m] -= lk * S[m*33 + k];
    }
    __syncthreads();
  }
}

// Forward substitution L x = r within one wave via lane shuffles.
// lane holds r[lane] on entry, x[lane] on exit.  L lower, ld=33 (read-only).
__device__ __forceinline__ float fwd_solve32(const float* L, float rv, int lane)
{
  float t = rv;
  for (int k = 0; k < 32; ++k) {
    float tk = __shfl(t, k, 32);
    float xk = tk / L[k*33 + k];
    if (lane == k)      t = xk;
    else if (lane > k)  t -= L[lane*33 + k] * xk;
  }
  return t;
}

// GRU cell + softmax head.  xbuf[112], hv[64] in LDS; updates hv and alphaOut.
__device__ void gru_alpha(const float* W_ih, const float* W_hh,
                          const float* b_ih, const float* b_hh,
                          const float* W_out, const float* b_out,
                          const float* xbuf, float* hv, float* giv, float* ghv,
                          float* alphaOut, float* lg, int tid)
{
  if (tid < 192) {
    float s = b_ih[tid];
    const float* wr = W_ih + tid*112;
    for (int i = 0; i < 112; ++i) s += wr[i]*xbuf[i];
    giv[tid] = s;
    float s2 = b_hh[tid];
    const float* wh = W_hh + tid*64;
    for (int i = 0; i < 64; ++i) s2 += wh[i]*hv[i];
    ghv[tid] = s2;
  }
  __syncthreads();
  if (tid < 64) {
    float r  = 1.0f/(1.0f + __expf(-(giv[tid]      + ghv[tid])));
    float zg = 1.0f/(1.0f + __expf(-(giv[64+tid]   + ghv[64+tid])));
    float n  = tanhf(giv[128+tid] + r*ghv[128+tid]);
    giv[tid] = (1.0f - zg)*n + zg*hv[tid];   // stage h_new
  }
  __syncthreads();
  if (tid < 64) hv[tid] = giv[tid];
  __syncthreads();
  if (tid < 8) {
    float s = b_out[tid];
    const float* wo = W_out + tid*64;
    for (int i = 0; i < 64; ++i) s += wo[i]*hv[i];
    lg[tid] = s;
  }
  __syncthreads();
  if (tid == 0) {
    float m = lg[0];
    for (int j = 1; j < 8; ++j) m = fmaxf(m, lg[j]);
    float e[8]; float se = 0.0f;
    for (int j = 0; j < 8; ++j) { e[j] = __expf(lg[j]-m); se += e[j]; }
    for (int j = 0; j < 8; ++j) alphaOut[j] = e[j]/se;
  }
  __syncthreads();
}

// ===========================================================================
__global__ void __launch_bounds__(256)
imm_kalman_kernel(const float* __restrict__ a_seq,  const float* __restrict__ h_obs,
                  const float* __restrict__ A_g,    const float* __restrict__ C_g,
                  const float* __restrict__ maskp,  const float* __restrict__ mat_Q,
                  const float* __restrict__ mat_R,  const float* __restrict__ P0,
                  const float* __restrict__ W_ih,   const float* __restrict__ W_hh,
                  const float* __restrict__ b_ih,   const float* __restrict__ b_hh,
                  const float* __restrict__ W_out,  const float* __restrict__ b_out,
                  float* __restrict__ out)
{
  extern __shared__ float sm[];
  const int tid  = threadIdx.x;
  const int wave = tid >> 5;
  const int lane = tid & 31;
  const int b    = blockIdx.x;

  float* P    = sm + oP;    float* Pf   = sm + oPf;   float* Amix = sm + oAmix;
  float* Tt   = sm + oTt;   float* IKC  = sm + oIKC;  float* Qm   = sm + oQm;
  float* Cmix = sm + oCmix; float* Cst  = sm + oCst;  float* PCm  = sm + oPCm;
  float* Kg   = sm + oKg;   float* Sall = sm + oSall; float* Sk   = sm + oSk;
  float* Linv = sm + oLinv; float* Sinv = sm + oSinv; float* Rm   = sm + oRm;
  float* Rall = sm + oRall; float* Av   = sm + oAv;   float* Zv   = sm + oZv;
  float* Zf   = sm + oZf;   float* Af   = sm + oAf;   float* Xb   = sm + oXb;
  float* Gi   = sm + oGi;   float* Gh   = sm + oGh;   float* Hv   = sm + oHv;
  float* Al   = sm + oAl;   float* Ll   = sm + oLl;   float* Lg   = sm + oLg;

  // ---------------- init: R, Q, P0, z0, h0 ----------------
  for (int idx = tid; idx < 1024; idx += 256) {
    int a = idx >> 5, c = idx & 31;
    float s = (a == c) ? 1e-4f : 0.0f;
    for (int k = 0; k < 32; ++k) s += mat_R[a*32+k]*mat_R[c*32+k];
    Rm[idx] = s;
  }
  for (int idx = tid; idx < 4096; idx += 256) {
    int a = idx >> 6, c = idx & 63;
    float s = (a == c) ? 1e-4f : 0.0f;
    for (int k = 0; k < 64; ++k) s += mat_Q[a*64+k]*mat_Q[c*64+k];
    Qm[idx] = s;
    P[idx]  = P0[idx];
  }
  if (tid < 32) {
    float a0 = a_seq[(size_t)b*Tc*DAc + tid];
    float a1 = a_seq[(size_t)b*Tc*DAc + DAc + tid];
    float vel = a1 - a0;
    Af[tid]      = a0 - vel;     // a_prev
    Zv[tid]      = a0 - vel;     // z0 low
    Zv[32 + tid] = vel;          // z0 high
  }
  if (tid < 64) Hv[tid] = 0.0f;
  __syncthreads();
  if (b == 0) {
    for (int idx = tid; idx < 1024; idx += 256) out[O_R + idx] = Rm[idx];
    for (int idx = tid; idx < 4096; idx += 256) out[O_Q + idx] = Qm[idx];
  }
  // initial alpha_net
  if (tid < 112) {
    float v;
    if (tid < 32)       v = Af[tid];
    else if (tid < 48)  v = h_obs[(size_t)b*DOBSc + (tid - 32)];
    else                v = Zv[tid - 48];
    Xb[tid] = v;
  }
  __syncthreads();
  gru_alpha(W_ih, W_hh, b_ih, b_hh, W_out, b_out, Xb, Hv, Gi, Gh, Al, Lg, tid);

  // ======================= time recursion =======================
  for (int t = 0; t < Tc; ++t) {
    const float mk = maskp[(size_t)b*Tc + t];
    if (tid < 32) Av[tid] = a_seq[((size_t)b*Tc + t)*DAc + tid];
    if (t + 1 < Tc) __builtin_prefetch(&a_seq[((size_t)b*Tc + t + 1)*DAc], 0, 1);
    __syncthreads();

    // ---------- IMM: per-model S_j = C_j P C_j^T + R, residual r_j ----------
    for (int j = 0; j < Kc; ++j) {
      for (int idx = tid; idx < 2048; idx += 256) Cst[idx] = C_g[j*2048 + idx];
      __syncthreads();
      if (tid < 32) {
        float s = Av[tid];
        const float* cr = Cst + tid*64;
        for (int i = 0; i < 64; ++i) s -= cr[i]*Zv[i];
        Rall[j*32 + tid] = s;
      }
      mm_wave(PCm, 32, P, 64, false, Cst, 64, true, 64, 32, 64, nullptr, 0, wave, lane);
      __syncthreads();
      mm_wave(Sall + j*1056, 33, Cst, 64, false, PCm, 32, false, 32, 32, 64, Rm, 32, wave, lane);
      __syncthreads();
    }
    // symmetrize + 1e-4 jitter
    for (int idx = tid; idx < 8192; idx += 256) {
      int j = idx >> 10, q = idx & 1023, i = q >> 5, c = q & 31;
      if (i <= c) {
        float* S = Sall + j*1056;
        float v = 0.5f*(S[i*33+c] + S[c*33+i]);
        if (i == c) v += 1e-4f;
        S[i*33+c] = v; S[c*33+i] = v;
      }
    }
    __syncthreads();
    chol32_block(Sall + wave*1056, lane);       // 8 models in parallel
    {
      const float* L = Sall + wave*1056;
      float y  = fwd_solve32(L, Rall[wave*32 + lane], lane);
      float ss = y*y;
      float ld = 2.0f*__logf(L[lane*33 + lane]);
      for (int off = 16; off; off >>= 1) {
        ss += __shfl_xor(ss, off, 32);
        ld += __shfl_xor(ld, off, 32);
      }
      if (lane == 0)
        Ll[wave] = -0.5f*(ss + 32.0f*1.8378770664093453f + ld);
    }
    __syncthreads();
    if (tid == 0) {
      float m = Ll[0];
      for (int j = 1; j < 8; ++j) m = fmaxf(m, Ll[j]);
      float e[8]; float se = 0.0f;
      for (int j = 0; j < 8; ++j) { e[j] = __expf(Ll[j]-m); se += e[j]; }
      for (int j = 0; j < 8; ++j)
        out[O_AIMM + ((size_t)b*Tc + t)*Kc + j] = (e[j]/se)*mk;
    }

    // ---------- mix A,C with carried alpha ----------
    float al[8];
    #pragma unroll
    for (int j = 0; j < 8; ++j) al[j] = Al[j];
    for (int idx = tid; idx < 4096; idx += 256) {
      float s = 0.0f;
      #pragma unroll
      for (int j = 0; j < 8; ++j) s += al[j]*A_g[j*4096 + idx];
      Amix[idx] = s;
    }
    for (int idx = tid; idx < 2048; idx += 256) {
      float s = 0.0f;
      #pragma unroll
      for (int j = 0; j < 8; ++j) s += al[j]*C_g[j*2048 + idx];
      Cmix[idx] = s;
    }
    __syncthreads();

    // ---------- Kalman update ----------
    if (tid < 32) {                       // r_k = a_k - Cmix z
      float s = Av[tid];
      const float* cr = Cmix + tid*64;
      for (int i = 0; i < 64; ++i) s -= cr[i]*Zv[i];
      Rall[tid] = s;
    }
    mm_wave(PCm, 32, P, 64, false, Cmix, 64, true, 64, 32, 64, nullptr, 0, wave, lane);
    __syncthreads();
    mm_wave(Sk, 33, Cmix, 64, false, PCm, 32, false, 32, 32, 64, Rm, 32, wave, lane);
    __syncthreads();
    for (int idx = tid; idx < 1024; idx += 256)   // S_pred out (pre-factor)
      out[O_S + ((size_t)b*Tc + t)*1024 + idx] = Sk[(idx>>5)*33 + (idx&31)];
    __syncthreads();
    chol32_block(wave == 0 ? Sk : Sall + wave*1056, lane);  // wave0 factors S_k
    // L^{-1} columns: 4 unit-vector solves per wave
    for (int c = wave; c < 32; c += 8) {
      float x = fwd_solve32(Sk, (lane == c) ? 1.0f : 0.0f, lane);
      Linv[lane*33 + c] = x;
    }
    __syncthreads();
    mm_wave(Sinv, 33, Linv, 33, true, Linv, 33, false, 32, 32, 32, nullptr, 0, wave, lane);
    __syncthreads();
    mm_wave(Kg, 32, PCm, 32, false, Sinv, 33, false, 64, 32, 32, nullptr, 0, wave, lane);
    __syncthreads();
    for (int idx = tid; idx < 2048; idx += 256) Kg[idx] *= mk;
    __syncthreads();
    if (tid < 64) {                       // z_filt = z + K r
      float s = Zv[tid];
      const float* kr = Kg + tid*32;
      for (int i = 0; i < 32; ++i) s += kr[i]*Rall[i];
      Zf[tid] = s;
      out[O_ZF + ((size_t)b*Tc + t)*64 + tid] = s;
    }
    __syncthreads();
    // IKC = I - K Cmix
    mm_wave(IKC, 64, Kg, 32, false, Cmix, 64, false, 64, 64, 32, nullptr, 0, wave, lane);
    __syncthreads();
    for (int idx = tid; idx < 4096; idx += 256) {
      int i = idx >> 6, c = idx & 63;
      float v = -IKC[idx];
      if (i == c) v += 1.0f;
      IKC[idx] = v;
    }
    __syncthreads();
    // P_filt = IKC P IKC^T + K R K^T
    mm_wave(Tt, 64, IKC, 64, false, P, 64, false, 64, 64, 64, nullptr, 0, wave, lane);
    __syncthreads();
    mm_wave(Pf, 64, Tt, 64, false, IKC, 64, true, 64, 64, 64, nullptr, 0, wave, lane);
    __syncthreads();
    mm_wave(PCm, 32, Kg, 32, false, Rm, 32, false, 64, 32, 32, nullptr, 0, wave, lane);
    __syncthreads();
    mm_wave(Pf, 64, PCm, 32, false, Kg, 32, true, 64, 64, 32, Pf, 64, wave, lane);
    __syncthreads();
    for (int idx = tid; idx < 4096; idx += 256) {  // symmetrize
      int i = idx >> 6, c = idx & 63;
      if (i <= c) {
        float v = 0.5f*(Pf[i*64+c] + Pf[c*64+i]);
        Pf[i*64+c] = v; Pf[c*64+i] = v;
      }
    }
    __syncthreads();
    for (int idx = tid; idx < 4096; idx += 256)
      out[O_PF + ((size_t)b*Tc + t)*4096 + idx] = Pf[idx];
    if (tid < 32) {                        // a_filt = Cmix z_filt
      float s = 0.0f;
      const float* cr = Cmix + tid*64;
      for (int i = 0; i < 64; ++i) s += cr[i]*Zf[i];
      Af[tid] = s;
      out[O_AF + ((size_t)b*Tc + t)*32 + tid] = s;
    }
    __syncthreads();

    // ---------- alpha_net ----------
    if (tid < 112) {
      float v;
      if (tid < 32)       v = Af[tid];
      else if (tid < 48)  v = h_obs[(size_t)b*DOBSc + (tid - 32)];
      else                v = Zf[tid - 48];
      Xb[tid] = v;
    }
    __syncthreads();
    gru_alpha(W_ih, W_hh, b_ih, b_hh, W_out, b_out, Xb, Hv, Gi, Gh, Al, Lg, tid);
    if (tid < 8) out[O_AL + ((size_t)b*Tc + t)*Kc + tid] = Al[tid];

    // ---------- prediction with alpha_new ----------
    #pragma unroll
    for (int j = 0; j < 8; ++j) al[j] = Al[j];
    for (int idx = tid; idx < 4096; idx += 256) {
      float s = 0.0f;
      #pragma unroll
      for (int j = 0; j < 8; ++j) s += al[j]*A_g[j*4096 + idx];
      Amix[idx] = s;
    }
    for (int idx = tid; idx < 2048; idx += 256) {
      float s = 0.0f;
      #pragma unroll
      for (int j = 0; j < 8; ++j) s += al[j]*C_g[j*2048 + idx];
      Cmix[idx] = s;
    }
    __syncthreads();
    if (tid < 64) {                        // z_pred = A2 z_filt (becomes carry)
      float s = 0.0f;
      const float* ar = Amix + tid*64;
      for (int i = 0; i < 64; ++i) s += ar[i]*Zf[i];
      Zv[tid] = s;
      out[O_ZP + ((size_t)b*Tc + t)*64 + tid] = s;
    }
    __syncthreads();
    // P_pred = A2 P_filt A2^T + Q  (written into P = next carry)
    mm_wave(Tt, 64, Amix, 64, false, Pf, 64, false, 64, 64, 64, nullptr, 0, wave, lane);
    __syncthreads();
    mm_wave(P, 64, Tt, 64, false, Amix, 64, true, 64, 64, 64, Qm, 64, wave, lane);
    __syncthreads();
    for (int idx = tid; idx < 4096; idx += 256) {
      int i = idx >> 6, c = idx & 63;
      if (i <= c) {
        float v = 0.5f*(P[i*64+c] + P[c*64+i]);
        P[i*64+c] = v; P[c*64+i] = v;
      }
    }
    __syncthreads();
    for (int idx = tid; idx < 4096; idx += 256)
      out[O_PP + ((size_t)b*Tc + t)*4096 + idx] = P[idx];
    if (tid < 32) {                        // a_pred = C2 z_pred
      float s = 0.0f;
      const float* cr = Cmix + tid*64;
      for (int i = 0; i < 64; ++i) s += cr[i]*Zv[i];
      out[O_AP + ((size_t)b*Tc + t)*32 + tid] = s;
    }
    __syncthreads();
  }
}

extern "C" void kernel_launch(void* const* d_in, const int* in_sizes, int n_in,
                              void* d_out, int out_size, void* d_ws, size_t ws_size,
                              hipStream_t stream)
{
  (void)in_sizes; (void)n_in; (void)out_size; (void)d_ws; (void)ws_size;
  const float* a_seq = (const float*)d_in[0];
  const float* h_obs = (const float*)d_in[1];
  const float* A_g   = (const float*)d_in[2];
  const float* C_g   = (const float*)d_in[3];
  const float* maskp = (const float*)d_in[4];
  const float* mat_Q = (const float*)d_in[5];
  const float* mat_R = (const float*)d_in[6];
  const float* P0    = (const float*)d_in[7];
  const float* W_ih  = (const float*)d_in[8];
  const float* W_hh  = (const float*)d_in[9];
  const float* b_ih  = (const float*)d_in[10];
  const float* b_hh  = (const float*)d_in[11];
  const float* W_out = (const float*)d_in[12];
  const float* b_out = (const float*)d_in[13];
  float* out = (float*)d_out;

  hipFuncSetAttribute((const void*)imm_kalman_kernel,
                      hipFuncAttributeMaxDynamicSharedMemorySize, SMEM_BYTES);
  imm_kalman_kernel<<<Bc, 256, SMEM_BYTES, stream>>>(
      a_seq, h_obs, A_g, C_g, maskp, mat_Q, mat_R, P0,
      W_ih, W_hh, b_ih, b_hh, W_out, b_out, out);
}